// constrained_network_65420941853144
// MI455X (gfx1250) — compile-verified
//
#include <hip/hip_runtime.h>
#include <math.h>

// Problem constants (match reference)
#define CS_   64
#define CV_   32
#define CG_   32
#define NB_   16
#define EMB_  8
#define NW_   192     // CS+CS+CV+CV
#define LAY_  3
#define ECHUNK 32768  // edges per radial-MLP chunk (multiple of 16); w-chunk = 25MB -> lives in L2

typedef float v2f __attribute__((ext_vector_type(2)));
typedef float v8f __attribute__((ext_vector_type(8)));

// ---------------------------------------------------------------------------
// semi_unitary: 10 Newton iterations on the 2x32 projection. One wave, lane=u.
// ---------------------------------------------------------------------------
__global__ void k_semi_unitary(const float* __restrict__ proj, float* __restrict__ Mout) {
  int u = threadIdx.x; // 32 lanes
  float m0 = proj[u], m1 = proj[32 + u];
  for (int it = 0; it < 10; ++it) {
    float g00 = m0 * m0, g01 = m0 * m1, g11 = m1 * m1;
    for (int o = 16; o > 0; o >>= 1) {
      g00 += __shfl_xor(g00, o);
      g01 += __shfl_xor(g01, o);
      g11 += __shfl_xor(g11, o);
    }
    float a00 = g00 - 1.0f, a01 = g01, a11 = g11 - 1.0f;
    float n0 = m0 - 0.5f * (a00 * m0 + a01 * m1);
    float n1 = m1 - 0.5f * (a01 * m0 + a11 * m1);
    m0 = n0; m1 = n1;
  }
  Mout[u] = m0;
  Mout[32 + u] = m1;
}

// ---------------------------------------------------------------------------
// Node init: v planes = x2 @ M, s = 0, z = embed[node_attr]. One block / node.
// v stored as 3 planes [nN x 32] (plane stride sv).
// ---------------------------------------------------------------------------
__global__ void k_init_nodes(const float* __restrict__ x, const int* __restrict__ node_attr,
                             const float* __restrict__ embed, const float* __restrict__ Mm,
                             float* __restrict__ s, float* __restrict__ v, size_t sv,
                             float* __restrict__ z, int nN) {
  int n = blockIdx.x;
  int t = threadIdx.x; // 64 threads
  if (n >= nN) return;
  if (t < 8) z[(size_t)n * 8 + t] = embed[(size_t)node_attr[n] * 8 + t];
  s[(size_t)n * 64 + t] = 0.0f;
  if (t < 32) {
    float mu0 = Mm[t], mu1 = Mm[32 + t];
    #pragma unroll
    for (int d = 0; d < 3; ++d) {
      float xv0 = x[(size_t)n * 6 + d];       // x2[n,0,d]
      float xv1 = x[(size_t)n * 6 + 3 + d];   // x2[n,1,d]
      v[(size_t)d * sv + (size_t)n * 32 + t] = xv0 * mu0 + xv1 * mu1;
    }
  }
}

// ---------------------------------------------------------------------------
// Edge geometry: Y1 (sqrt3 * dhat) and Gaussian radial basis rb[16].
// ---------------------------------------------------------------------------
__global__ void k_edge_geom(const float* __restrict__ x, const int* __restrict__ esrc,
                            const int* __restrict__ edst, float* __restrict__ Y1,
                            float* __restrict__ rb, int nE) {
  int e = blockIdx.x * blockDim.x + threadIdx.x;
  if (e >= nE) return;
  int s_ = esrc[e], d_ = edst[e];
  float dx = x[(size_t)s_ * 6 + 3] - x[(size_t)d_ * 6 + 3];
  float dy = x[(size_t)s_ * 6 + 4] - x[(size_t)d_ * 6 + 4];
  float dz = x[(size_t)s_ * 6 + 5] - x[(size_t)d_ * 6 + 5];
  float r = sqrtf(dx * dx + dy * dy + dz * dz + 1e-12f);
  float inv = 1.0f / r;
  const float SQ3 = 1.7320508075688772f;
  Y1[(size_t)e * 3 + 0] = SQ3 * dx * inv;
  Y1[(size_t)e * 3 + 1] = SQ3 * dy * inv;
  Y1[(size_t)e * 3 + 2] = SQ3 * dz * inv;
  const float INVW = 16.0f / 5.0f;        // 1/width, width = MAX_R/NB
  const float STEP = 5.0f / 15.0f;        // linspace(0,5,16) spacing
  #pragma unroll
  for (int j = 0; j < NB_; ++j) {
    float t = (r - STEP * (float)j) * INVW;
    rb[(size_t)e * NB_ + j] = __expf(-t * t) * 4.0f; // * sqrt(NB)
  }
}

// ---------------------------------------------------------------------------
// Register-blocked f32 WMMA GEMM: D[MxN] = act(A[MxK] @ B[KxN] + C + bias)
//   - one wave (32 threads / block) computes NT adjacent 16x16 output tiles
//   - A fragment loaded ONCE per k-step and reused across NT WMMAs
//   - requires N % (16*NT) == 0 ; K % 4 == 0 (all call sites satisfy this)
//   - C == nullptr -> zero accumulator; D may alias C (tile read-then-write)
//   - act: 0 = none, 1 = silu
// A fragment (16x4 f32): lanes 0-15 rows M, VGPR0/1 = K0/K1; lanes 16-31 K2/K3.
// B fragment (4x16 f32): VGPR0 lanes0-15 row K0 / lanes16-31 row K2; VGPR1 K1/K3.
// C/D (16x16 f32): VGPR i, lanes0-15 M=i, lanes16-31 M=i+8; N = lane&15.
// ---------------------------------------------------------------------------
template <int NT>
__global__ void k_gemm_wmma(const float* __restrict__ A, const float* __restrict__ B,
                            const float* __restrict__ C, float* __restrict__ D,
                            const float* __restrict__ bias,
                            int M, int N, int K, int act) {
  const int lane = threadIdx.x & 31;
  const int hi   = lane >> 4;
  const int l15  = lane & 15;
  const int m0   = blockIdx.x * 16;
  const int n0   = blockIdx.y * (16 * NT);

  v8f acc[NT];
  #pragma unroll
  for (int t = 0; t < NT; ++t) {
    #pragma unroll
    for (int i = 0; i < 8; ++i) {
      int m = m0 + i + 8 * hi;
      acc[t][i] = (C && m < M) ? C[(size_t)m * N + n0 + t * 16 + l15] : 0.0f;
    }
  }

  // loop-invariant base pointers (strength-reduced addressing in the k-loop)
  const float* Ap = A + (size_t)(m0 + l15) * K + 2 * hi;         // + k0
  const float* Bp = B + (size_t)(2 * hi) * N + n0 + l15;         // + k0*N (+t*16)

  for (int k0 = 0; k0 < K; k0 += 4) {
    v2f a = *(const v2f*)(Ap + k0);
    #pragma unroll
    for (int t = 0; t < NT; ++t) {
      v2f b;
      b.x = Bp[(size_t)k0 * N + t * 16];
      b.y = Bp[(size_t)(k0 + 1) * N + t * 16];
      acc[t] = __builtin_amdgcn_wmma_f32_16x16x4_f32(false, a, false, b, (short)0,
                                                     acc[t], false, false);
    }
  }

  #pragma unroll
  for (int t = 0; t < NT; ++t) {
    #pragma unroll
    for (int i = 0; i < 8; ++i) {
      int m = m0 + i + 8 * hi, n = n0 + t * 16 + l15;
      float xv = acc[t][i];
      if (bias) xv += bias[n];
      if (act == 1) xv = xv * (1.0f / (1.0f + __expf(-xv)));  // silu
      if (m < M) D[(size_t)m * N + n] = xv;
    }
  }
}

// ---------------------------------------------------------------------------
// Outer-product builds for the z-contracted self-connections.
// ys[n, c*8+a] = s[n,c]*z[n,a]          (nN x 512)
// yv[d][n, u*8+a] = v[d][n,u]*z[n,a]    (3 planes, nN x 256)
// ---------------------------------------------------------------------------
__global__ void k_build_ys(const float* __restrict__ s, const float* __restrict__ z,
                           float* __restrict__ ys, int nN) {
  size_t i = (size_t)blockIdx.x * blockDim.x + threadIdx.x;
  size_t tot = (size_t)nN * 512;
  if (i >= tot) return;
  int n = (int)(i >> 9);
  int c = (int)((i >> 3) & 63);
  int a = (int)(i & 7);
  ys[i] = s[(size_t)n * 64 + c] * z[(size_t)n * 8 + a];
}

__global__ void k_build_yv(const float* __restrict__ v, size_t sv,
                           const float* __restrict__ z,
                           float* __restrict__ yv, size_t syv, int nN) {
  size_t i = (size_t)blockIdx.x * blockDim.x + threadIdx.x;
  size_t per = (size_t)nN * 256;
  if (i >= 3 * per) return;
  int d = (int)(i / per);
  size_t r = i - (size_t)d * per;
  int n = (int)(r >> 8);
  int u = (int)((r >> 3) & 31);
  int a = (int)(r & 7);
  yv[(size_t)d * syv + r] = v[(size_t)d * sv + (size_t)n * 32 + u] * z[(size_t)n * 8 + a];
}

// ---------------------------------------------------------------------------
// Per-edge message formation + scatter-add.  One wave per edge (lane = channel).
// w chunk-local; edge index offset e0 into Y1/esrc/edst.
// ---------------------------------------------------------------------------
__global__ void k_message(const float* __restrict__ w,
                          const float* __restrict__ s1,
                          const float* __restrict__ v1, size_t sv,
                          const float* __restrict__ Y1,
                          const int* __restrict__ esrc, const int* __restrict__ edst,
                          float* __restrict__ aggs,
                          float* __restrict__ aggv, size_t sav,
                          int e0, int cnt) {
  const int lane = threadIdx.x & 31;
  const int el = blockIdx.x * 8 + (threadIdx.x >> 5);
  if (el >= cnt) return;
  const int e = e0 + el;
  const float INV_SQRT_NEIGH = 0.17677669529663687f; // 1/sqrt(32)
  const float INV_SQRT3 = 0.5773502691896258f;
  const int src = esrc[e];
  const int dst = edst[e];
  const float y0 = Y1[(size_t)e * 3 + 0];
  const float y1 = Y1[(size_t)e * 3 + 1];
  const float y2 = Y1[(size_t)e * 3 + 2];
  const float* we = w + (size_t)el * NW_;

  // vector channels (u = lane)
  const float vs0 = v1[0 * sv + (size_t)src * 32 + lane];
  const float vs1 = v1[1 * sv + (size_t)src * 32 + lane];
  const float vs2 = v1[2 * sv + (size_t)src * 32 + lane];
  const float dot = (vs0 * y0 + vs1 * y1 + vs2 * y2) * INV_SQRT3;
  atomicAdd(&aggs[(size_t)dst * 96 + 64 + lane], we[160 + lane] * dot * INV_SQRT_NEIGH);
  {
    const float t = we[128 + lane] * INV_SQRT_NEIGH; // w_vv
    atomicAdd(&aggv[0 * sav + (size_t)dst * 96 + 64 + lane], t * vs0);
    atomicAdd(&aggv[1 * sav + (size_t)dst * 96 + 64 + lane], t * vs1);
    atomicAdd(&aggv[2 * sav + (size_t)dst * 96 + 64 + lane], t * vs2);
  }
  // scalar channels (c = lane, lane+32)
  #pragma unroll
  for (int h = 0; h < 2; ++h) {
    const int c = lane + 32 * h;
    const float ss = s1[(size_t)src * 64 + c];
    atomicAdd(&aggs[(size_t)dst * 96 + c], we[c] * ss * INV_SQRT_NEIGH); // w_ss * ss
    const float t = we[64 + c] * ss * INV_SQRT_NEIGH;                    // w_sv * ss
    atomicAdd(&aggv[0 * sav + (size_t)dst * 96 + c], t * y0);
    atomicAdd(&aggv[1 * sav + (size_t)dst * 96 + c], t * y1);
    atomicAdd(&aggv[2 * sav + (size_t)dst * 96 + c], t * y2);
  }
}

// ---------------------------------------------------------------------------
// Gating: gs = silu(out_s[:, :64]); gv[d] = sigmoid(out_s[:, 64+u]) * out_v[d]
// ---------------------------------------------------------------------------
__global__ void k_gate(const float* __restrict__ out_s,
                       const float* __restrict__ out_v, size_t sov,
                       float* __restrict__ gs,
                       float* __restrict__ gv, size_t sgv, int nN) {
  size_t i = (size_t)blockIdx.x * blockDim.x + threadIdx.x;
  size_t tot = (size_t)nN * 96;
  if (i >= tot) return;
  int n = (int)(i / 96);
  int c = (int)(i % 96);
  float xv = out_s[i];
  if (c < 64) {
    gs[(size_t)n * 64 + c] = xv * (1.0f / (1.0f + __expf(-xv)));
  } else {
    int u = c - 64;
    float g = 1.0f / (1.0f + __expf(-xv));
    #pragma unroll
    for (int d = 0; d < 3; ++d)
      gv[(size_t)d * sgv + (size_t)n * 32 + u] =
          g * out_v[(size_t)d * sov + (size_t)n * 32 + u];
  }
}

// ---------------------------------------------------------------------------
// Output: x_out[n, i*3+d] = sum_u v[d][n,u] * M[i,u]
// ---------------------------------------------------------------------------
__global__ void k_output(const float* __restrict__ v, size_t sv,
                         const float* __restrict__ Mm, float* __restrict__ out, int nN) {
  size_t i = (size_t)blockIdx.x * blockDim.x + threadIdx.x;
  if (i >= (size_t)nN * 6) return;
  int n = (int)(i / 6);
  int k = (int)(i % 6);
  int ii = k / 3, d = k % 3;
  float acc = 0.0f;
  const float* vp = v + (size_t)d * sv + (size_t)n * 32;
  const float* mp = Mm + (size_t)ii * 32;
  #pragma unroll
  for (int u = 0; u < 32; ++u) acc += vp[u] * mp[u];
  out[i] = acc;
}

// ---------------------------------------------------------------------------
// Host side
// ---------------------------------------------------------------------------
static inline void gemm(const float* A, const float* B, const float* C, float* D,
                        const float* bias, int M, int N, int K, int act, hipStream_t st) {
  dim3 blk(32, 1, 1);                   // one wave per block
  int mg = (M + 15) / 16;
  // pick largest NT in {4,3,2,1} with N % (16*NT) == 0
  if (N % 64 == 0) {
    dim3 grd(mg, N / 64, 1);
    hipLaunchKernelGGL(k_gemm_wmma<4>, grd, blk, 0, st, A, B, C, D, bias, M, N, K, act);
  } else if (N % 48 == 0) {
    dim3 grd(mg, N / 48, 1);
    hipLaunchKernelGGL(k_gemm_wmma<3>, grd, blk, 0, st, A, B, C, D, bias, M, N, K, act);
  } else if (N % 32 == 0) {
    dim3 grd(mg, N / 32, 1);
    hipLaunchKernelGGL(k_gemm_wmma<2>, grd, blk, 0, st, A, B, C, D, bias, M, N, K, act);
  } else {
    dim3 grd(mg, (N + 15) / 16, 1);
    hipLaunchKernelGGL(k_gemm_wmma<1>, grd, blk, 0, st, A, B, C, D, bias, M, N, K, act);
  }
}

extern "C" void kernel_launch(void* const* d_in, const int* in_sizes, int n_in,
                              void* d_out, int out_size, void* d_ws, size_t ws_size,
                              hipStream_t stream) {
  const float* x        = (const float*)d_in[0];
  const int*   nattr    = (const int*)d_in[2];
  const int*   esrc     = (const int*)d_in[3];
  const int*   edst     = (const int*)d_in[4];
  const float* proj     = (const float*)d_in[5];
  const float* embed    = (const float*)d_in[6];
  const float* Wr1      = (const float*)d_in[7];
  const float* br1      = (const float*)d_in[8];
  const float* Wr2      = (const float*)d_in[9];
  const float* Wsc_s    = (const float*)d_in[10];
  const float* Wsc_v    = (const float*)d_in[11];
  const float* W1_s     = (const float*)d_in[12];
  const float* W1_v     = (const float*)d_in[13];
  const float* W2_s     = (const float*)d_in[14];
  const float* W2_v     = (const float*)d_in[15];
  const float* Wsi_s    = (const float*)d_in[16];
  const float* Wsi_v    = (const float*)d_in[17];

  const int nN = in_sizes[2];   // 10000
  const int nE = in_sizes[3];   // 320000

  // --- workspace carve-up (floats) ---
  float* p = (float*)d_ws;
  auto alloc = [&](size_t cnt) { float* r = p; p += cnt; return r; };
  const size_t SV  = (size_t)nN * 32;   // v / v1 / gv plane stride
  const size_t SAV = (size_t)nN * 96;   // agg_v plane stride
  const size_t SYV = (size_t)nN * 256;  // yv plane stride

  float* Mm     = alloc(64);
  float* z      = alloc((size_t)nN * 8);
  float* s      = alloc((size_t)nN * 64);
  float* v      = alloc(3 * SV);
  float* Y1     = alloc((size_t)nE * 3);
  float* rb     = alloc((size_t)nE * NB_);
  float* s1     = alloc((size_t)nN * 64);
  float* v1     = alloc(3 * SV);
  float* ys     = alloc((size_t)nN * 512);
  float* yv     = alloc(3 * SYV);
  float* sc_s   = alloc((size_t)nN * 96);
  float* sc_v   = alloc(3 * SV);
  float* agg_s  = alloc((size_t)nN * 96);
  float* agg_v  = alloc(3 * SAV);
  float* out_s  = alloc((size_t)nN * 96);
  float* out_v  = alloc(3 * SV);
  float* gs     = alloc((size_t)nN * 64);
  float* gv     = alloc(3 * SV);
  float* hid    = alloc((size_t)ECHUNK * 64);   // radial MLP chunk buffers
  float* wchunk = alloc((size_t)ECHUNK * NW_);
  (void)ws_size;

  // --- setup ---
  hipLaunchKernelGGL(k_semi_unitary, dim3(1), dim3(32), 0, stream, proj, Mm);
  hipLaunchKernelGGL(k_init_nodes, dim3(nN), dim3(64), 0, stream,
                     x, nattr, embed, Mm, s, v, SV, z, nN);
  hipLaunchKernelGGL(k_edge_geom, dim3((nE + 255) / 256), dim3(256), 0, stream,
                     x, esrc, edst, Y1, rb, nE);

  for (int l = 0; l < LAY_; ++l) {
    const float* Wr1l  = Wr1   + (size_t)l * NB_ * 64;
    const float* br1l  = br1   + (size_t)l * 64;
    const float* Wr2l  = Wr2   + (size_t)l * 64 * NW_;
    const float* Wscsl = Wsc_s + (size_t)l * 512 * 96;
    const float* Wscvl = Wsc_v + (size_t)l * 256 * 32;
    const float* W1sl  = W1_s  + (size_t)l * 64 * 64;
    const float* W1vl  = W1_v  + (size_t)l * 32 * 32;
    const float* W2sl  = W2_s  + (size_t)l * 96 * 96;
    const float* W2vl  = W2_v  + (size_t)l * 96 * 32;
    const float* Wsisl = Wsi_s + (size_t)l * 64 * 64;
    const float* Wsivl = Wsi_v + (size_t)l * 32 * 32;

    // self-connection contractions (z outer products -> GEMM)
    {
      size_t tot = (size_t)nN * 512;
      hipLaunchKernelGGL(k_build_ys, dim3((tot + 255) / 256), dim3(256), 0, stream,
                         s, z, ys, nN);
      gemm(ys, Wscsl, nullptr, sc_s, nullptr, nN, 96, 512, 0, stream);
      size_t totv = 3 * (size_t)nN * 256;
      hipLaunchKernelGGL(k_build_yv, dim3((totv + 255) / 256), dim3(256), 0, stream,
                         v, SV, z, yv, SYV, nN);
      for (int d = 0; d < 3; ++d)
        gemm(yv + (size_t)d * SYV, Wscvl, nullptr, sc_v + (size_t)d * SV, nullptr,
             nN, 32, 256, 0, stream);
    }

    // node pre-linear: s1 = s@W1_s, v1[d] = v[d]@W1_v
    gemm(s, W1sl, nullptr, s1, nullptr, nN, 64, 64, 0, stream);
    for (int d = 0; d < 3; ++d)
      gemm(v + (size_t)d * SV, W1vl, nullptr, v1 + (size_t)d * SV, nullptr,
           nN, 32, 32, 0, stream);

    // zero aggregation targets
    hipMemsetAsync(agg_s, 0, (size_t)nN * 96 * sizeof(float), stream);
    hipMemsetAsync(agg_v, 0, 3 * SAV * sizeof(float), stream);

    // chunked radial MLP (w-chunk stays in L2) + fused message/scatter consume
    for (int e0 = 0; e0 < nE; e0 += ECHUNK) {
      int cnt = nE - e0; if (cnt > ECHUNK) cnt = ECHUNK;
      gemm(rb + (size_t)e0 * NB_, Wr1l, nullptr, hid, br1l, cnt, 64, NB_, 1, stream);
      gemm(hid, Wr2l, nullptr, wchunk, nullptr, cnt, NW_, 64, 0, stream);
      hipLaunchKernelGGL(k_message, dim3((cnt + 7) / 8), dim3(256), 0, stream,
                         wchunk, s1, v1, SV, Y1, esrc, edst,
                         agg_s, agg_v, SAV, e0, cnt);
    }

    // out_s = sc_s + agg_s @ W2_s ; out_v[d] = sc_v[d] + agg_v[d] @ W2_v
    gemm(agg_s, W2sl, sc_s, out_s, nullptr, nN, 96, 96, 0, stream);
    for (int d = 0; d < 3; ++d)
      gemm(agg_v + (size_t)d * SAV, W2vl, sc_v + (size_t)d * SV,
           out_v + (size_t)d * SV, nullptr, nN, 32, 96, 0, stream);

    // gating
    {
      size_t tot = (size_t)nN * 96;
      hipLaunchKernelGGL(k_gate, dim3((tot + 255) / 256), dim3(256), 0, stream,
                         out_s, out_v, SV, gs, gv, SV, nN);
    }

    // residual updates: s += gs@Wsi_s ; v[d] += gv[d]@Wsi_v   (C aliases D: safe)
    gemm(gs, Wsisl, s, s, nullptr, nN, 64, 64, 0, stream);
    for (int d = 0; d < 3; ++d)
      gemm(gv + (size_t)d * SV, Wsivl, v + (size_t)d * SV, v + (size_t)d * SV,
           nullptr, nN, 32, 32, 0, stream);
  }

  // final projection back: x_out[n, i*3+d] = sum_u v[d][n,u]*M[i,u]
  {
    size_t tot = (size_t)nN * 6;
    hipLaunchKernelGGL(k_output, dim3((tot + 255) / 256), dim3(256), 0, stream,
                       v, SV, Mm, (float*)d_out, nN);
  }
}